// Model_79542794322454
// MI455X (gfx1250) — compile-verified
//
#include <hip/hip_runtime.h>
#include <cstdint>

// Problem sizes fixed by the reference setup_inputs().
#define CTC_B 32
#define CTC_T 1024
#define CTC_C 801          // blank = 800
#define CTC_L 128
#define CTC_S (2 * CTC_L + 1)   // 257 extended states
#define TB    288               // 9 wave32s, covers S=257
#define NEGV  (-1.0e30f)

// ---------------------------------------------------------------------------
// CDNA5 async global->LDS copy (ASYNCcnt-tracked) helpers.
// global_load_async_to_lds_b32  vdst(LDS byte addr), vaddr(64-bit global), off
// ---------------------------------------------------------------------------
__device__ __forceinline__ void async_g2l_b32(uint32_t lds_byte_addr, const float* gptr) {
  uint64_t ga = (uint64_t)(uintptr_t)gptr;
  asm volatile("global_load_async_to_lds_b32 %0, %1, off"
               :: "v"(lds_byte_addr), "v"(ga)
               : "memory");
}
__device__ __forceinline__ void wait_async_le6() {
  asm volatile("s_wait_asynccnt 0x6" ::: "memory");
}
__device__ __forceinline__ void wait_async_le3() {
  asm volatile("s_wait_asynccnt 0x3" ::: "memory");
}
__device__ __forceinline__ void wait_async_0() {
  asm volatile("s_wait_asynccnt 0x0" ::: "memory");
}

// ---------------------------------------------------------------------------
// Kernel 1: per-(b,t) softmax log-denominator. One wave32 per row of C=801.
// Single streaming pass (online max/sum), wave shuffle reduction. HBM-bound:
// 105 MB once ~= 4.5 us at 23.3 TB/s. Dataset < 192 MB global L2, so this
// pass also leaves the logits L2-resident for kernel 2.
// ---------------------------------------------------------------------------
__global__ __launch_bounds__(256) void ctc_denom_kernel(const float* __restrict__ logits,
                                                        float* __restrict__ denom) {
  const int lane = threadIdx.x & 31;
  const int wave = threadIdx.x >> 5;
  const int r = blockIdx.x * 8 + wave;            // row index in [0, B*T)
  const float* p = logits + (size_t)r * CTC_C;

  float m = -3.402823466e38f, s = 0.0f;           // online softmax state
  for (int i = lane; i < CTC_C; i += 32) {
    float x = p[i];
    float mn = fmaxf(m, x);
    s = s * expf(m - mn) + expf(x - mn);
    m = mn;
  }
  // wave32 reduction (warpSize == 32 on gfx1250)
  for (int off = 16; off > 0; off >>= 1) {
    float mo = __shfl_xor(m, off, 32);
    float so = __shfl_xor(s, off, 32);
    float mn = fmaxf(m, mo);
    s = s * expf(m - mn) + so * expf(mo - mn);
    m = mn;
  }
  if (lane == 0) denom[r] = m + logf(s);
}

// ---------------------------------------------------------------------------
// Kernel 2: CTC forward recursion. One workgroup per batch element,
// thread-per-extended-state, alpha double-buffered in LDS, ONE barrier per
// time step. Logits rows are 4-way buffered in LDS via async global->LDS
// copies issued TWO steps ahead; with in-order ASYNCcnt retirement,
// s_wait_asynccnt<=6 leaves at most rows t+1/t+2 pending, so row t is
// guaranteed resident. Own alpha value is carried in a register; the alpha
// arrays have a 2-entry NEG pad so s-1/s-2 reads are branch-free.
// ---------------------------------------------------------------------------
__global__ __launch_bounds__(TB) void ctc_alpha_kernel(const float* __restrict__ logits,
                                                       const int* __restrict__ labels,
                                                       const int* __restrict__ lens,
                                                       const float* __restrict__ denom,
                                                       float* __restrict__ losses) {
  __shared__ float row[4][CTC_C];       // staged logits rows (quad buffer)
  __shared__ float alpha[2][TB + 2];    // 2-entry NEG pad at the front
  __shared__ float sden[CTC_T];         // per-t log-denominators for this batch

  const int b = blockIdx.x;
  const int s = threadIdx.x;
  const int idx = s + 2;                // position inside padded alpha rows
  const int blank = CTC_C - 1;
  const int ll = lens[b];
  const bool active = (s < CTC_S);
  const bool valid  = active && (s < 2 * ll + 1);

  // Extended-label symbol and skip-transition permission for this state.
  int ext = blank;
  bool allow2 = false;
  if (active && (s & 1)) {
    ext = labels[b * CTC_L + (s >> 1)];
    int prev2 = (s >= 3) ? labels[b * CTC_L + ((s - 2) >> 1)] : -1;
    allow2 = (s >= 2) && (ext != blank) && (ext != prev2);
  }

  const float* lbase = logits + (size_t)b * CTC_T * CTC_C;
  const float* dbase = denom + (size_t)b * CTC_T;

  // Preload all T denominators (4 KB) into LDS; init the alpha front pads.
  for (int i = s; i < CTC_T; i += TB) sden[i] = dbase[i];
  if (s < 2) { alpha[0][s] = NEGV; alpha[1][s] = NEGV; }

  auto copy_row = [&](int t, int buf) {
    const float* src = lbase + (size_t)t * CTC_C;
#pragma unroll
    for (int k = 0; k < 3; ++k) {
      int i = s + k * TB;
      if (i >= CTC_C) i = CTC_C - 1;  // clamped duplicate loads: no divergence
      async_g2l_b32((uint32_t)(uintptr_t)&row[buf][i], src + i);
    }
  };

  // Prologue: stage rows 0..2; <=6 pending means row 0 is complete.
  copy_row(0, 0);
  copy_row(1, 1);
  copy_row(2, 2);
  wait_async_le6();
  __syncthreads();

  // t = 0 initialization (matches reference: base + emit, then validity mask).
  float a_self;
  {
    float e0 = row[0][ext] - sden[0];
    float a;
    if (s == 0)      a = e0;
    else if (s == 1) a = ((ll > 0) ? 0.0f : NEGV) + e0;
    else             a = NEGV + e0;
    if (!valid) a = NEGV;
    a_self = active ? a : NEGV;
    alpha[0][idx] = a_self;
  }
  int cur = 0;

  for (int t = 1; t < CTC_T; ++t) {
    // Issue row t+2; guarantee row t resident (in-order ASYNCcnt retirement).
    if (t + 2 < CTC_T)      { copy_row(t + 2, (t + 2) & 3); wait_async_le6(); }
    else if (t + 1 < CTC_T) { wait_async_le3(); }
    else                    { wait_async_0(); }
    __syncthreads();   // row[t&3] and alpha[cur] now visible to all waves

    const float* r = row[t & 3];
    float emit = r[ext] - sden[t];
    float a0 = a_self;
    float a1 = alpha[cur][idx - 1];
    float a2 = allow2 ? alpha[cur][idx - 2] : NEGV;
    float m = fmaxf(a0, fmaxf(a1, a2));
    float v = m + __logf(__expf(a0 - m) + __expf(a1 - m) + __expf(a2 - m)) + emit;
    if (!valid) v = NEGV;
    a_self = active ? v : NEGV;
    alpha[cur ^ 1][idx] = a_self;
    cur ^= 1;
  }
  __syncthreads();

  if (s == 0) {
    float e1 = alpha[cur][2 * ll + 2];
    float e2 = (ll > 0) ? alpha[cur][2 * ll + 1] : NEGV;
    float m = fmaxf(e1, e2);
    losses[b] = -(m + logf(expf(e1 - m) + expf(e2 - m)));
  }
}

// ---------------------------------------------------------------------------
// Kernel 3: deterministic mean over the B=32 per-batch losses (one wave32).
// ---------------------------------------------------------------------------
__global__ void ctc_finish_kernel(const float* __restrict__ losses,
                                  float* __restrict__ out, int B) {
  float v = (threadIdx.x < B) ? losses[threadIdx.x] : 0.0f;
  for (int off = 16; off > 0; off >>= 1) v += __shfl_xor(v, off, 32);
  if (threadIdx.x == 0) out[0] = v / (float)B;
}

extern "C" void kernel_launch(void* const* d_in, const int* in_sizes, int n_in,
                              void* d_out, int out_size, void* d_ws, size_t ws_size,
                              hipStream_t stream) {
  (void)in_sizes; (void)n_in; (void)out_size; (void)ws_size;
  const float* logits = (const float*)d_in[0];
  const int*   labels = (const int*)d_in[1];
  const int*   lens   = (const int*)d_in[2];

  float* denom  = (float*)d_ws;                     // B*T floats
  float* losses = denom + (size_t)CTC_B * CTC_T;    // B floats

  ctc_denom_kernel<<<(CTC_B * CTC_T) / 8, 256, 0, stream>>>(logits, denom);
  ctc_alpha_kernel<<<CTC_B, TB, 0, stream>>>(logits, labels, lens, denom, losses);
  ctc_finish_kernel<<<1, 32, 0, stream>>>(losses, (float*)d_out, CTC_B);
}